// PermopNestedTensors_26345329393705
// MI455X (gfx1250) — compile-verified
//
#include <hip/hip_runtime.h>

// ---------------------------------------------------------------------------
// Segment-sum as one-hot GEMM on CDNA5 WMMA:
//   OUT[16 x 256] = ONEHOT[16 x T] * VALUES[T x 256]
// using V_WMMA_F32_16X16X4_F32 (D = A(16x4 f32) * B(4x16 f32) + C(16x16 f32)).
// Memory-bound problem (~134 MB @ 23.3 TB/s ~ 5.8 us floor); WMMA is just the
// cheapest correct way to do the scatter-accumulate while streaming.
// ---------------------------------------------------------------------------

typedef __attribute__((ext_vector_type(2))) float v2f;
typedef __attribute__((ext_vector_type(8))) float v8f;

#define DCOLS 256          // feature dim
#define NSEG  16           // batch segments (matches WMMA M=16)
#define WAVES_PER_BLOCK 16 // 16 waves x 16 cols = 256 columns per block
#define BLOCK_THREADS (WAVES_PER_BLOCK * 32)
#define MAX_PARTIALS 256   // partial tiles in workspace (4 MB @ 256)

// One WMMA step: consume 4 tokens starting at tbase for this wave's 16-col tile.
// A-matrix 16x4 layout: lane l -> M = l&15, K = 2*(l>>4) + {0,1} (2 VGPRs)
// B-matrix 4x16 layout: lane l -> K = 2*(l>>4) + {0,1}, N = l&15 (2 VGPRs)
__device__ __forceinline__ v8f wmma_step(const float* __restrict__ values,
                                         const int*   __restrict__ segids,
                                         int tbase, int half, int mrow, int col,
                                         v8f acc) {
    const int krow = tbase + (half << 1);
    // segment ids for this lane's two K slots (8B aligned: krow is even)
    const int2 ids = *(const int2*)(segids + krow);
    v2f a;
    a.x = (ids.x == mrow) ? 1.0f : 0.0f;
    a.y = (ids.y == mrow) ? 1.0f : 0.0f;
    v2f b;
    b.x = values[(size_t)krow * DCOLS + col];
    b.y = values[((size_t)krow + 1) * DCOLS + col];
    // 8 args: (neg_a, A, neg_b, B, c_mod, C, reuse_a, reuse_b)
    return __builtin_amdgcn_wmma_f32_16x16x4_f32(false, a, false, b,
                                                 (short)0, acc, false, false);
}

__global__ __launch_bounds__(BLOCK_THREADS)
void seg_sum_wmma_partials(const float* __restrict__ values,
                           const int*   __restrict__ segids,
                           float*       __restrict__ partials,
                           int total_tokens) {
    const int tid  = threadIdx.x;
    const int lane = tid & 31;
    const int wave = tid >> 5;
    const int half = lane >> 4;            // 0 or 1 (which 16-lane group)
    const int mrow = lane & 15;            // A row / B,C column within tile
    const int col  = wave * 16 + mrow;     // global column 0..255

    const int chunks = total_tokens >> 2;  // full 4-token chunks (tail done later)
    const int stride = gridDim.x;

    v8f acc0 = {};
    v8f acc1 = {};

    int c = blockIdx.x;
    // 2-way unrolled grid-stride loop; two independent C chains hide WMMA latency.
    for (; c + stride < chunks; c += 2 * stride) {
        acc0 = wmma_step(values, segids, c << 2,            half, mrow, col, acc0);
        acc1 = wmma_step(values, segids, (c + stride) << 2, half, mrow, col, acc1);
    }
    for (; c < chunks; c += stride) {
        acc0 = wmma_step(values, segids, c << 2, half, mrow, col, acc0);
    }
    v8f acc = acc0 + acc1;

    // C 16x16 layout: lane l holds rows m = v + 8*half (v=0..7) at column col.
    float* p = partials + (size_t)blockIdx.x * (NSEG * DCOLS);
#pragma unroll
    for (int v = 0; v < 8; ++v) {
        const int m = v + 8 * half;
        p[m * DCOLS + col] = acc[v];
    }
}

// Deterministic fixed-order reduction of the per-block partial tiles,
// plus scalar handling of a (total % 4) token tail for robustness.
__global__ __launch_bounds__(256)
void reduce_partials(const float* __restrict__ partials,
                     const float* __restrict__ values,
                     const int*   __restrict__ segids,
                     float*       __restrict__ out,
                     int nb, int tail_start, int total_tokens) {
    const int i    = blockIdx.x * blockDim.x + threadIdx.x;  // 0..4095
    const int m    = i >> 8;                                 // segment
    const int dcol = i & (DCOLS - 1);                        // column
    float s = 0.0f;
    for (int p = 0; p < nb; ++p)
        s += partials[(size_t)p * (NSEG * DCOLS) + i];
    for (int t = tail_start; t < total_tokens; ++t)
        if (segids[t] == m)
            s += values[(size_t)t * DCOLS + dcol];
    out[i] = s;
}

extern "C" void kernel_launch(void* const* d_in, const int* in_sizes, int n_in,
                              void* d_out, int out_size, void* d_ws, size_t ws_size,
                              hipStream_t stream) {
    const float* values = (const float*)d_in[0];
    const int*   segids = (const int*)d_in[1];
    float*       out    = (float*)d_out;
    float*       parts  = (float*)d_ws;

    const int total = in_sizes[1];          // number of tokens (131072)

    // Pick the partial-tile count: largest power of two <= MAX_PARTIALS that
    // fits in the workspace (each tile is 16*256*4 = 16 KB).
    int nb = MAX_PARTIALS;
    const size_t tile_bytes = (size_t)NSEG * DCOLS * sizeof(float);
    while (nb > 1 && (size_t)nb * tile_bytes > ws_size) nb >>= 1;

    seg_sum_wmma_partials<<<nb, BLOCK_THREADS, 0, stream>>>(values, segids, parts, total);

    const int tail_start = total & ~3;
    reduce_partials<<<(NSEG * DCOLS) / 256, 256, 0, stream>>>(
        parts, values, segids, out, nb, tail_start, total);
}